// VNSmall_26525718020207
// MI455X (gfx1250) — compile-verified
//
#include <hip/hip_runtime.h>
#include <math.h>

// ---------------------------------------------------------------------------
// VN "graph feature cross" block for MI455X (gfx1250, wave32).
//
// Stage 1 (dominant): KNN via V_WMMA_F32_16X16X4_F32 pairwise dot products
//   score(n,m) = |p_m|^2 - 2 p_n.p_m   (query-norm dropped: rank-invariant)
//   unsorted top-50 per query kept in LDS (replace-max + rescan) -- the set
//   of neighbors is all downstream math needs (mean over k).
// Stage 2: per-(b,n,k) norms -> per-block partial sums (deterministic).
// Stage 3: finalize BN mean / inv-std (single block, fixed reduction order).
// Stage 4: recompute q,d; BN-rescale; directional projection; per-block
//          partial sums of out[o][v].
// Stage 5: reduce partials -> d_out (B,3,3) = 36 floats.
// ---------------------------------------------------------------------------

#define NPTS   8192
#define KNN    50
#define QPB    128        // queries per KNN block = 4 waves
#define CCH    16         // candidate chunk (WMMA tile N)
#define ST     18         // stage row stride (pad: avoids LDS bank conflicts)
#define EPSF   1e-6f
#define BNEPS  1e-5f
#define NBATCH 4
#define ITEMS_PER_BATCH (NPTS * KNN)            // 409600
#define BLK_PER_BATCH   (ITEMS_PER_BATCH / 256) // 1600
#define NBLK_ITEMS      (NBATCH * BLK_PER_BATCH)// 6400

typedef __attribute__((ext_vector_type(2))) float v2f;
typedef __attribute__((ext_vector_type(8))) float v8f;

// ---------------------------------------------------------------------------
// Stage 1: KNN.  grid = (NPTS/QPB, NBATCH), block = QPB threads (4 waves).
// Each wave owns 32 queries; per 16-candidate chunk it issues two
// v_wmma_f32_16x16x4_f32 ops (queries 0-15 and 16-31 of the wave) and
// scatters the 16x16 dot tiles into LDS for the scalar top-k pass.
//
// ISA VGPR layouts assumed (CDNA5 ISA §7.12.2):
//   A 16x4 f32 : lanes 0-15 -> {K0,K1}=(x,y), lanes 16-31 -> {K2,K3}=(z,0)
//   B 4x16 f32 : lanes 0-15 (N=lane) -> {K0,K1}, lanes 16-31 -> {K2,K3}
//   D 16x16 f32: n = lane%16, m = r + 8*(lane>=16)
// ---------------------------------------------------------------------------
__global__ __launch_bounds__(QPB) void knn_kernel(const float* __restrict__ x,
                                                  int* __restrict__ nbr_idx)
{
    __shared__ float stage[QPB * ST];
    __shared__ float kd[KNN * QPB];     // top-k scores, k-major (stride QPB)
    __shared__ int   ki[KNN * QPB];     // top-k indices
    __shared__ float candx[CCH], candy[CCH], candz[CCH], candn[CCH];

    const int b     = blockIdx.y;
    const int qbase = blockIdx.x * QPB;
    const int t     = threadIdx.x;
    const int lane  = t & 31;
    const int wave  = t >> 5;
    const int half  = lane >> 4;        // 0: K=0,1 lanes   1: K=2,3 lanes
    const int m     = lane & 15;        // M index (A) / N index (B,D)
    const float* __restrict__ xb = x + (size_t)b * NPTS * 3;

    for (int j = 0; j < KNN; ++j) kd[j * QPB + t] = 1e30f;
    float cur_max = 1e30f;
    int   cur_arg = 0;

    // Loop-invariant A operands (this wave's 32 query points).
    const int q_lo = qbase + wave * 32 + m;
    const int q_hi = q_lo + 16;
    v2f a_lo, a_hi;
    if (half == 0) {
        a_lo.x = xb[q_lo * 3 + 0]; a_lo.y = xb[q_lo * 3 + 1];
        a_hi.x = xb[q_hi * 3 + 0]; a_hi.y = xb[q_hi * 3 + 1];
    } else {
        a_lo.x = xb[q_lo * 3 + 2]; a_lo.y = 0.0f;
        a_hi.x = xb[q_hi * 3 + 2]; a_hi.y = 0.0f;
    }

    for (int c0 = 0; c0 < NPTS; c0 += CCH) {
        __syncthreads();                       // stage/cand reuse fence
        if (t < CCH) {
            const float px = xb[(c0 + t) * 3 + 0];
            const float py = xb[(c0 + t) * 3 + 1];
            const float pz = xb[(c0 + t) * 3 + 2];
            candx[t] = px; candy[t] = py; candz[t] = pz;
            candn[t] = px * px + py * py + pz * pz;
        }
        __syncthreads();

        // B operand: 16 candidate points, zero-padded K=3.
        v2f bop;
        if (half == 0) { bop.x = candx[m]; bop.y = candy[m]; }
        else           { bop.x = candz[m]; bop.y = 0.0f;     }

        v8f acc_lo = {};
        acc_lo = __builtin_amdgcn_wmma_f32_16x16x4_f32(
            false, a_lo, false, bop, (short)0, acc_lo, false, false);
        v8f acc_hi = {};
        acc_hi = __builtin_amdgcn_wmma_f32_16x16x4_f32(
            false, a_hi, false, bop, (short)0, acc_hi, false, false);

        // Scatter dot tiles: stage[q_local][n], row stride 18 (conflict-free).
#pragma unroll
        for (int r = 0; r < 8; ++r) {
            const int mrow = r + 8 * half;
            stage[(wave * 32 + mrow)      * ST + m] = acc_lo[r];
            stage[(wave * 32 + 16 + mrow) * ST + m] = acc_hi[r];
        }
        __syncthreads();

        // Top-k maintenance: thread t == query (qbase + t), column t of kd/ki.
        for (int j = 0; j < CCH; ++j) {
            const float s = candn[j] - 2.0f * stage[t * ST + j];
            if (s < cur_max) {
                kd[cur_arg * QPB + t] = s;
                ki[cur_arg * QPB + t] = c0 + j;
                float mx = -1e30f; int am = 0;
                for (int u = 0; u < KNN; ++u) {
                    const float v = kd[u * QPB + t];
                    if (v > mx) { mx = v; am = u; }
                }
                cur_max = mx; cur_arg = am;
            }
        }
    }

    int* __restrict__ op = nbr_idx + ((size_t)b * NPTS + qbase) * KNN;
    for (int j = 0; j < KNN; ++j)
        op[(size_t)t * KNN + j] = ki[j * QPB + t];
}

// ---------------------------------------------------------------------------
// Shared helpers for the per-(b,n,k) item kernels.
// ---------------------------------------------------------------------------
__device__ __forceinline__ void load_feat(const float* __restrict__ xb,
                                          int n, int nb, float f[3][3])
{
    const float cx = xb[n  * 3 + 0], cy = xb[n  * 3 + 1], cz = xb[n  * 3 + 2];
    const float px = xb[nb * 3 + 0], py = xb[nb * 3 + 1], pz = xb[nb * 3 + 2];
    f[0][0] = px - cx; f[0][1] = py - cy; f[0][2] = pz - cz;   // edge
    f[1][0] = cx;      f[1][1] = cy;      f[1][2] = cz;        // center
    f[2][0] = py * cz - pz * cy;                               // cross(n, c)
    f[2][1] = pz * cx - px * cz;
    f[2][2] = px * cy - py * cx;
}

__device__ __forceinline__ void block_reduce_store(float val, float* red,
                                                   float* dst, int t)
{
    __syncthreads();
    red[t] = val;
    __syncthreads();
    for (int off = 128; off > 0; off >>= 1) {
        if (t < off) red[t] += red[t + off];
        __syncthreads();
    }
    if (t == 0) *dst = red[0];
}

// ---------------------------------------------------------------------------
// Stage 2: per-item norms -> per-block partial {sum, sumsq} per channel.
// ---------------------------------------------------------------------------
__global__ __launch_bounds__(256) void stats_kernel(const float* __restrict__ x,
                                                    const int* __restrict__ nbr_idx,
                                                    const float* __restrict__ Wf,
                                                    float* __restrict__ partial)
{
    __shared__ float red[256];
    const int t    = threadIdx.x;
    const int b    = blockIdx.x / BLK_PER_BATCH;
    const int item = (blockIdx.x % BLK_PER_BATCH) * 256 + t;
    const int n    = item / KNN;
    const int kk   = item % KNN;
    const float* __restrict__ xb = x + (size_t)b * NPTS * 3;
    const int nb = nbr_idx[((size_t)b * NPTS + n) * KNN + kk];

    float f[3][3];
    load_feat(xb, n, nb, f);

    float vals[6];
#pragma unroll
    for (int o = 0; o < 3; ++o) {
        const float w0 = Wf[o * 3 + 0], w1 = Wf[o * 3 + 1], w2 = Wf[o * 3 + 2];
        const float q0 = w0 * f[0][0] + w1 * f[1][0] + w2 * f[2][0];
        const float q1 = w0 * f[0][1] + w1 * f[1][1] + w2 * f[2][1];
        const float q2 = w0 * f[0][2] + w1 * f[1][2] + w2 * f[2][2];
        const float nrm = sqrtf(q0 * q0 + q1 * q1 + q2 * q2) + EPSF;
        vals[o]     = nrm;
        vals[3 + o] = nrm * nrm;
    }
    for (int i = 0; i < 6; ++i)
        block_reduce_store(vals[i], red, &partial[(size_t)blockIdx.x * 6 + i], t);
}

// ---------------------------------------------------------------------------
// Stage 3: finalize BN stats (deterministic fixed-order reduction).
// stats[0..2] = mean_o, stats[3..5] = rsqrt(var_o + BN_EPS)
// ---------------------------------------------------------------------------
__global__ __launch_bounds__(256) void stats_finalize_kernel(
        const float* __restrict__ partial, float* __restrict__ stats)
{
    __shared__ float red[256];
    __shared__ float colsum[6];
    const int t = threadIdx.x;
    for (int i = 0; i < 6; ++i) {
        float a = 0.0f;
        for (int blk = t; blk < NBLK_ITEMS; blk += 256)
            a += partial[(size_t)blk * 6 + i];
        __syncthreads();
        red[t] = a;
        __syncthreads();
        for (int off = 128; off > 0; off >>= 1) {
            if (t < off) red[t] += red[t + off];
            __syncthreads();
        }
        if (t == 0) colsum[i] = red[0];
    }
    __syncthreads();
    if (t == 0) {
        const float inv_cnt = 1.0f / (float)((size_t)NBATCH * NPTS * KNN);
        for (int o = 0; o < 3; ++o) {
            const float mean = colsum[o] * inv_cnt;
            const float var  = colsum[3 + o] * inv_cnt - mean * mean;
            stats[o]     = mean;
            stats[3 + o] = rsqrtf(var + BNEPS);
        }
    }
}

// ---------------------------------------------------------------------------
// Stage 4: recompute q,d; BN rescale; directional leaky projection
// (NEG_SLOPE = 0); per-block partials of out[o][v].
// ---------------------------------------------------------------------------
__global__ __launch_bounds__(256) void out_kernel(const float* __restrict__ x,
                                                  const int* __restrict__ nbr_idx,
                                                  const float* __restrict__ Wf,
                                                  const float* __restrict__ Wd,
                                                  const float* __restrict__ gamma,
                                                  const float* __restrict__ beta,
                                                  const float* __restrict__ stats,
                                                  float* __restrict__ partial)
{
    __shared__ float red[256];
    const int t    = threadIdx.x;
    const int b    = blockIdx.x / BLK_PER_BATCH;
    const int item = (blockIdx.x % BLK_PER_BATCH) * 256 + t;
    const int n    = item / KNN;
    const int kk   = item % KNN;
    const float* __restrict__ xb = x + (size_t)b * NPTS * 3;
    const int nb = nbr_idx[((size_t)b * NPTS + n) * KNN + kk];

    float f[3][3];
    load_feat(xb, n, nb, f);

    float acc[9];
#pragma unroll
    for (int o = 0; o < 3; ++o) {
        const float w0 = Wf[o * 3 + 0], w1 = Wf[o * 3 + 1], w2 = Wf[o * 3 + 2];
        float q0 = w0 * f[0][0] + w1 * f[1][0] + w2 * f[2][0];
        float q1 = w0 * f[0][1] + w1 * f[1][1] + w2 * f[2][1];
        float q2 = w0 * f[0][2] + w1 * f[1][2] + w2 * f[2][2];
        const float u0 = Wd[o * 3 + 0], u1 = Wd[o * 3 + 1], u2 = Wd[o * 3 + 2];
        const float d0 = u0 * f[0][0] + u1 * f[1][0] + u2 * f[2][0];
        const float d1 = u0 * f[0][1] + u1 * f[1][1] + u2 * f[2][1];
        const float d2 = u0 * f[0][2] + u1 * f[1][2] + u2 * f[2][2];

        const float nrm = sqrtf(q0 * q0 + q1 * q1 + q2 * q2) + EPSF;
        const float nbn = (nrm - stats[o]) * stats[3 + o] * gamma[o] + beta[o];
        const float ratio = nbn / nrm;
        q0 *= ratio; q1 *= ratio; q2 *= ratio;

        const float dot  = q0 * d0 + q1 * d1 + q2 * d2;
        const float dsq  = d0 * d0 + d1 * d1 + d2 * d2;
        const float corr = (dot >= 0.0f) ? 0.0f : dot / (dsq + EPSF);
        acc[o * 3 + 0] = q0 - corr * d0;
        acc[o * 3 + 1] = q1 - corr * d1;
        acc[o * 3 + 2] = q2 - corr * d2;
    }
    for (int i = 0; i < 9; ++i)
        block_reduce_store(acc[i], red, &partial[(size_t)blockIdx.x * 9 + i], t);
}

// ---------------------------------------------------------------------------
// Stage 5: reduce per-block partials -> d_out (B,3,3), divide by N*K.
// ---------------------------------------------------------------------------
__global__ __launch_bounds__(64) void final_reduce_kernel(
        const float* __restrict__ partial, float* __restrict__ out)
{
    const int t = threadIdx.x;
    if (t < NBATCH * 9) {
        const int b = t / 9, i = t % 9;
        float a = 0.0f;
        for (int blk = 0; blk < BLK_PER_BATCH; ++blk)
            a += partial[((size_t)(b * BLK_PER_BATCH + blk)) * 9 + i];
        out[t] = a * (1.0f / (float)(NPTS * KNN));
    }
}

// ---------------------------------------------------------------------------
extern "C" void kernel_launch(void* const* d_in, const int* in_sizes, int n_in,
                              void* d_out, int out_size, void* d_ws, size_t ws_size,
                              hipStream_t stream)
{
    (void)in_sizes; (void)n_in; (void)out_size; (void)ws_size;

    const float* x     = (const float*)d_in[0];   // (B, N, 3)
    const float* Wf    = (const float*)d_in[1];   // (3, 3)
    const float* Wd    = (const float*)d_in[2];   // (3, 3)
    const float* gamma = (const float*)d_in[3];   // (3,)
    const float* beta  = (const float*)d_in[4];   // (3,)
    float* out = (float*)d_out;                   // (B, 3, 3) = 36 floats

    // Workspace layout (~7 MB total).
    char*  ws       = (char*)d_ws;
    int*   nbr_idx  = (int*)ws;                                        // B*N*K ints
    float* partialB = (float*)(ws + (size_t)NBATCH * NPTS * KNN * sizeof(int));
    float* stats    = partialB + (size_t)NBLK_ITEMS * 6;
    float* partialD = stats + 8;

    knn_kernel<<<dim3(NPTS / QPB, NBATCH), QPB, 0, stream>>>(x, nbr_idx);
    stats_kernel<<<NBLK_ITEMS, 256, 0, stream>>>(x, nbr_idx, Wf, partialB);
    stats_finalize_kernel<<<1, 256, 0, stream>>>(partialB, stats);
    out_kernel<<<NBLK_ITEMS, 256, 0, stream>>>(x, nbr_idx, Wf, Wd, gamma, beta,
                                               stats, partialD);
    final_reduce_kernel<<<1, 64, 0, stream>>>(partialD, out);
}